// GCN_5454608466091
// MI455X (gfx1250) — compile-verified
//
#include <hip/hip_runtime.h>

// GCN forward for MI455X (gfx1250, wave32).
// Memory-bound workload (~5 GB traffic -> ~0.22 ms @ 23.3 TB/s); GEMM part
// (4.9 GFLOP) done with V_WMMA_F32_16X16X4_F32 to keep exact f32 semantics.

typedef __attribute__((ext_vector_type(2))) float v2f;
typedef __attribute__((ext_vector_type(8))) float v8f;

// ---------------------------------------------------------------- degree / norm
__global__ __launch_bounds__(256) void deg_init_kernel(float* __restrict__ deg, int n) {
  int i = blockIdx.x * 256 + threadIdx.x;
  if (i < n) deg[i] = 1.0f;                       // self-loop contribution
}

__global__ __launch_bounds__(256) void deg_count_kernel(const long long* __restrict__ col,
                                                        float* __restrict__ deg, int e) {
  int i = blockIdx.x * 256 + threadIdx.x;
  if (i < e) atomicAdd(&deg[(int)col[i]], 1.0f);  // in-degree at targets
}

__global__ __launch_bounds__(256) void deg_rsqrt_kernel(float* __restrict__ deg, int n) {
  int i = blockIdx.x * 256 + threadIdx.x;
  if (i < n) { float d = deg[i]; deg[i] = (d > 0.0f) ? rsqrtf(d) : 0.0f; }
}

// ---------------------------------------------------------------- dense transform
// Out[N,128] = X[N,128] @ W[128,128], f32.
// One block = 16 node rows x 128 features. 8 waves, each wave owns one 16x16
// output tile and runs a K-loop of 32 v_wmma_f32_16x16x4_f32.
// A-frag layout (16x4 f32): lanes 0-15 -> M=lane, K=k0/k0+1 in v[0]/v[1];
// lanes 16-31 -> M=lane-16, K=k0+2/k0+3.  B-frag (4x16) mirrored on N.
// C/D: 8 VGPRs, lanes 0-15 -> M=r, N=lane; lanes 16-31 -> M=r+8.
__global__ __launch_bounds__(256) void gemm128_kernel(const float* __restrict__ X,
                                                      const float* __restrict__ W,
                                                      float* __restrict__ Out, int n) {
  __shared__ float lx[16 * 132];                  // pad 128 -> 132 to dodge bank conflicts
  const int tid = threadIdx.x;
  const int m0  = blockIdx.x * 16;

  for (int i = tid; i < 16 * 128; i += 256) {     // stage 16x128 X tile to LDS
    int r = i >> 7, k = i & 127;
    int m = m0 + r;
    lx[r * 132 + k] = (m < n) ? X[(size_t)m * 128 + k] : 0.0f;
  }
  __syncthreads();

  const int wave = tid >> 5;                      // 8 waves -> 8 feature tiles
  const int lane = tid & 31;
  const int half = lane >> 4;                     // 0: K pair 0/1, 1: K pair 2/3
  const int l    = lane & 15;
  const int n0   = wave * 16;

  v8f c = {};
  #pragma unroll 4
  for (int kk = 0; kk < 32; ++kk) {
    const int ka = kk * 4 + half * 2;
    v2f a, b;
    a.x = lx[l * 132 + ka];
    a.y = lx[l * 132 + ka + 1];
    b.x = W[(size_t)ka * 128 + n0 + l];
    b.y = W[(size_t)(ka + 1) * 128 + n0 + l];
    c = __builtin_amdgcn_wmma_f32_16x16x4_f32(false, a, false, b,
                                              (short)0, c, false, false);
  }

  // Epilogue: block-uniform full-tile test -> scalar branch, unguarded stores
  // on the hot path (N % 16 == 0 means this is always taken for N = 50000).
  if (m0 + 16 <= n) {
    #pragma unroll
    for (int r = 0; r < 8; ++r)
      Out[(size_t)(m0 + r + half * 8) * 128 + n0 + l] = c[r];
  } else {
    #pragma unroll
    for (int r = 0; r < 8; ++r) {
      int m = m0 + r + half * 8;
      if (m < n) Out[(size_t)m * 128 + n0 + l] = c[r];
    }
  }
}

// ---------------------------------------------------------------- aggregation
// agg[v,:] = T[v,:] * dinv[v]^2 + bias   (self-loop + bias, fused init)
__global__ __launch_bounds__(256) void agg_init_kernel(const float4* __restrict__ T,
                                                       const float* __restrict__ dinv,
                                                       const float4* __restrict__ bias,
                                                       float4* __restrict__ A, int n) {
  int i = blockIdx.x * 256 + threadIdx.x;         // over n*32 float4 chunks
  if (i < n * 32) {
    int v = i >> 5, f4 = i & 31;
    float d = dinv[v];
    float w = d * d;
    float4 t = T[i];
    float4 b = bias[f4];
    A[i] = make_float4(fmaf(t.x, w, b.x), fmaf(t.y, w, b.y),
                       fmaf(t.z, w, b.z), fmaf(t.w, w, b.w));
  }
}

// One wave per edge; lane = float4 chunk of the 128-wide row.
// eid is forced wave-uniform via readfirstlane so the edge indices and dinv
// factors lower to scalar loads; the vector path is only the coalesced 512B
// row gather + 4 f32 global atomics per lane (no-return -> STOREcnt).
__global__ __launch_bounds__(256) void scatter_kernel(const float* __restrict__ T,
                                                      const long long* __restrict__ row,
                                                      const long long* __restrict__ col,
                                                      const float* __restrict__ dinv,
                                                      float* __restrict__ A, int e) {
  int t   = blockIdx.x * 256 + threadIdx.x;       // e*32 <= 51.2M, fits in int
  int eid = __builtin_amdgcn_readfirstlane(t >> 5);   // wave-uniform edge id
  int c4  = threadIdx.x & 31;                     // float4 chunk within row
  if (eid < e) {
    int r = (int)row[eid];
    int c = (int)col[eid];
    float w  = dinv[r] * dinv[c];
    float4 h = ((const float4*)(T + (size_t)r * 128))[c4];
    float* dst = A + (size_t)c * 128 + (size_t)c4 * 4;
    atomicAdd(dst + 0, h.x * w);
    atomicAdd(dst + 1, h.y * w);
    atomicAdd(dst + 2, h.z * w);
    atomicAdd(dst + 3, h.w * w);
  }
}

// ---------------------------------------------------------------- readout
__global__ void zero_sums_kernel(float* __restrict__ sums) {
  sums[threadIdx.x] = 0.0f;                       // <<<1,128>>>
}

__global__ __launch_bounds__(256) void col_reduce_kernel(const float* __restrict__ A,
                                                         float* __restrict__ sums, int n) {
  int f   = threadIdx.x & 127;                    // coalesced over features
  int sub = threadIdx.x >> 7;
  float acc = 0.0f;
  for (int v = blockIdx.x * 2 + sub; v < n; v += gridDim.x * 2)
    acc += A[(size_t)v * 128 + f];
  atomicAdd(&sums[f], acc);
}

__global__ void final_kernel(const float* __restrict__ sums,
                             const float* __restrict__ lw,
                             const float* __restrict__ lb,
                             float* __restrict__ out, float invN) {
  __shared__ float s[128];
  int f = threadIdx.x;                            // <<<1,128>>>
  s[f] = sums[f] * invN * lw[f];
  __syncthreads();
  for (int off = 64; off > 0; off >>= 1) {
    if (f < off) s[f] += s[f + off];
    __syncthreads();
  }
  if (f == 0) out[0] = s[0] + lb[0];
}

// ---------------------------------------------------------------- launcher
extern "C" void kernel_launch(void* const* d_in, const int* in_sizes, int n_in,
                              void* d_out, int out_size, void* d_ws, size_t ws_size,
                              hipStream_t stream) {
  (void)n_in; (void)out_size; (void)ws_size;
  const float*     x     = (const float*)d_in[0];
  const long long* ei    = (const long long*)d_in[1];   // int64 [2,E]
  const float*     W0    = (const float*)d_in[2];
  const float*     b0    = (const float*)d_in[3];
  const float*     W1    = (const float*)d_in[4];
  const float*     b1    = (const float*)d_in[5];
  const float*     W2    = (const float*)d_in[6];
  const float*     b2    = (const float*)d_in[7];
  const float*     lin_w = (const float*)d_in[8];
  const float*     lin_b = (const float*)d_in[9];

  const int N = in_sizes[0] / 128;
  const int E = in_sizes[1] / 2;
  const long long* row = ei;       // source nodes
  const long long* col = ei + E;   // target nodes

  // workspace carve-out (all 256B aligned): dinv[N] | sums[128] | T | A | B
  char* ws = (char*)d_ws;
  size_t off = 0;
  float* dinv = (float*)(ws + off); off += (((size_t)N * 4) + 255) & ~(size_t)255;
  float* sums = (float*)(ws + off); off += 512;
  const size_t featBytes = ((((size_t)N * 128 * 4) + 255) & ~(size_t)255);
  float* T = (float*)(ws + off); off += featBytes;
  float* A = (float*)(ws + off); off += featBytes;
  float* B = (float*)(ws + off); off += featBytes;

  const int mtiles = (N + 15) / 16;
  const int nThr   = N * 32;                  // float4 chunks over [N,128]
  const int eThr32 = E * 32;                  // one wave per edge

  // gcn_norm
  deg_init_kernel <<<(N + 255) / 256, 256, 0, stream>>>(dinv, N);
  deg_count_kernel<<<(E + 255) / 256, 256, 0, stream>>>(col, dinv, E);
  deg_rsqrt_kernel<<<(N + 255) / 256, 256, 0, stream>>>(dinv, N);

  // layer 0: T = x@W0 ; A = dinv^2*T + b0 + scatter
  gemm128_kernel <<<mtiles, 256, 0, stream>>>(x, W0, T, N);
  agg_init_kernel<<<(nThr + 255) / 256, 256, 0, stream>>>((const float4*)T, dinv,
                                                          (const float4*)b0, (float4*)A, N);
  scatter_kernel <<<(eThr32 + 255) / 256, 256, 0, stream>>>(T, row, col, dinv, A, E);

  // layer 1: T = A@W1 ; B = ...
  gemm128_kernel <<<mtiles, 256, 0, stream>>>(A, W1, T, N);
  agg_init_kernel<<<(nThr + 255) / 256, 256, 0, stream>>>((const float4*)T, dinv,
                                                          (const float4*)b1, (float4*)B, N);
  scatter_kernel <<<(eThr32 + 255) / 256, 256, 0, stream>>>(T, row, col, dinv, B, E);

  // layer 2: T = B@W2 ; A = ...
  gemm128_kernel <<<mtiles, 256, 0, stream>>>(B, W2, T, N);
  agg_init_kernel<<<(nThr + 255) / 256, 256, 0, stream>>>((const float4*)T, dinv,
                                                          (const float4*)b2, (float4*)A, N);
  scatter_kernel <<<(eThr32 + 255) / 256, 256, 0, stream>>>(T, row, col, dinv, A, E);

  // mean pool + linear head
  zero_sums_kernel <<<1, 128, 0, stream>>>(sums);
  col_reduce_kernel<<<256, 256, 0, stream>>>(A, sums, N);
  final_kernel     <<<1, 128, 0, stream>>>(sums, lin_w, lin_b, (float*)d_out,
                                           1.0f / (float)N);
}